// GCN_46437186404377
// MI455X (gfx1250) — compile-verified
//
#include <hip/hip_runtime.h>
#include <math.h>

typedef float v2f __attribute__((ext_vector_type(2)));
typedef float v8f __attribute__((ext_vector_type(8)));

#define N_NODES 50000
#define N_EDGES 800000
#define NFEAT   512
#define NHID    64
#define NCLASS  40

// ---------------------------------------------------------------------------
// GEMM1: XW = x[50000x512] @ W1[512x64]  via V_WMMA_F32_16X16X4_F32.
// W1 is staged once per block into LDS in a paired-K layout:
//   lds[p*128 + (n ^ ((p&1)<<4))*2 + j] = W1[2p + j][n]   (p = K/2, j = K&1)
// so each B fragment {W1[k][n], W1[k+1][n]} is ONE 8B ds_load_b64, and the
// XOR-16 column swizzle puts lane-halves (p vs p+1) on disjoint dword banks.
// A frag (16x4 f32): lanes 0-15 hold K=0,1 ; lanes 16-31 hold K=2,3.
// D (16x16 f32): element r -> (M = r + 8*(lane>>4), N = lane&15).
// ---------------------------------------------------------------------------
__global__ __launch_bounds__(256) void gemm1_wmma(const float* __restrict__ x,
                                                  const float* __restrict__ w1,
                                                  float* __restrict__ out) {
  extern __shared__ float lw[];                    // 256 pairs * 128 = 32768 f32 (128 KB)

  // ---- cooperative stage of W1 (8192 float4 loads across 256 threads) ----
  const float4* w4 = (const float4*)w1;
  for (int i = threadIdx.x; i < (NFEAT * NHID) / 4; i += blockDim.x) {
    const float4 v = w4[i];
    const int k = i >> 4;                          // 16 float4 per 64-wide row
    const int n = (i & 15) * 4;
    const int p = k >> 1, j = k & 1;
    float* dst = &lw[p * 128 + ((n ^ ((p & 1) << 4)) * 2) + j];
    dst[0] = v.x; dst[2] = v.y; dst[4] = v.z; dst[6] = v.w;
  }
  __syncthreads();

  const int wave = blockIdx.x * (blockDim.x >> 5) + (threadIdx.x >> 5);
  if (wave >= (N_NODES / 16)) return;              // wave-uniform exit (after barrier)
  const int lane = threadIdx.x & 31;
  const int half = lane >> 4;
  const int mrow = lane & 15;
  const int rowBase = wave * 16;

  // swizzled, loop-invariant column offsets for the 4 N-tiles
  int colsw[4];
#pragma unroll
  for (int nt = 0; nt < 4; ++nt)
    colsw[nt] = ((nt * 16 + mrow) ^ (half << 4)) * 2;

  const float* arow = x + (rowBase + mrow) * NFEAT + 2 * half;
  v8f acc0 = {}, acc1 = {}, acc2 = {}, acc3 = {};

  for (int k0 = 0; k0 < NFEAT / 4; ++k0) {
    const v2f a = *(const v2f*)(arow + k0 * 4);
    const float* base = &lw[(k0 * 2 + half) * 128];
    const v2f b0 = *(const v2f*)(base + colsw[0]);
    const v2f b1 = *(const v2f*)(base + colsw[1]);
    const v2f b2 = *(const v2f*)(base + colsw[2]);
    const v2f b3 = *(const v2f*)(base + colsw[3]);
    acc0 = __builtin_amdgcn_wmma_f32_16x16x4_f32(false, a, false, b0, (short)0, acc0, false, false);
    acc1 = __builtin_amdgcn_wmma_f32_16x16x4_f32(false, a, false, b1, (short)0, acc1, false, false);
    acc2 = __builtin_amdgcn_wmma_f32_16x16x4_f32(false, a, false, b2, (short)0, acc2, false, false);
    acc3 = __builtin_amdgcn_wmma_f32_16x16x4_f32(false, a, false, b3, (short)0, acc3, false, false);
  }

  float* orow = out + (rowBase + 8 * half) * NHID + mrow;
#pragma unroll
  for (int r = 0; r < 8; ++r) {
    orow[r * NHID + 0]  = acc0[r];
    orow[r * NHID + 16] = acc1[r];
    orow[r * NHID + 32] = acc2[r];
    orow[r * NHID + 48] = acc3[r];
  }
}

// ---------------------------------------------------------------------------
// GEMM2 (fused): HW = relu(acc1 + b1) @ W2   (N padded 40 -> 48, masked)
// W2 staged zero-padded in LDS pair layout (row stride 96 f32 self-rotates
// banks by 32 dwords between the two lane-halves -> conflict-free); b1 staged
// so each bias pair is one ds_load_b64 and relu is applied on the A fragment.
// ---------------------------------------------------------------------------
__global__ __launch_bounds__(256) void gemm2_wmma(const float* __restrict__ acc1,
                                                  const float* __restrict__ b1,
                                                  const float* __restrict__ w2,
                                                  float* __restrict__ out) {
  __shared__ float w2s[(NHID / 2) * 96];           // 32 pairs * 48 cols * 2 = 3072 f32
  __shared__ float b1s[NHID];

  for (int i = threadIdx.x; i < (NHID / 2) * 96; i += blockDim.x) {
    const int p = i / 96, r = i % 96;
    const int n = r >> 1, j = r & 1;
    w2s[i] = (n < NCLASS) ? w2[(2 * p + j) * NCLASS + n] : 0.f;
  }
  if (threadIdx.x < NHID) b1s[threadIdx.x] = b1[threadIdx.x];
  __syncthreads();

  const int wave = blockIdx.x * (blockDim.x >> 5) + (threadIdx.x >> 5);
  if (wave >= (N_NODES / 16)) return;
  const int lane = threadIdx.x & 31;
  const int half = lane >> 4;
  const int mrow = lane & 15;
  const int rowBase = wave * 16;
  const bool c2ok = (32 + mrow) < NCLASS;

  const float* arow = acc1 + (rowBase + mrow) * NHID + 2 * half;
  v8f acc0 = {}, acc1v = {}, acc2v = {};

  for (int k0 = 0; k0 < NHID / 4; ++k0) {
    const int k = k0 * 4 + 2 * half;
    const v2f ain = *(const v2f*)(arow + k0 * 4);
    const v2f bb  = *(const v2f*)&b1s[k];
    v2f a;
    a.x = fmaxf(ain.x + bb.x, 0.f);
    a.y = fmaxf(ain.y + bb.y, 0.f);
    const float* base = &w2s[(k0 * 2 + half) * 96 + mrow * 2];
    const v2f b0 = *(const v2f*)(base + 0);
    const v2f b1f = *(const v2f*)(base + 32);
    const v2f b2f = *(const v2f*)(base + 64);
    acc0  = __builtin_amdgcn_wmma_f32_16x16x4_f32(false, a, false, b0,  (short)0, acc0,  false, false);
    acc1v = __builtin_amdgcn_wmma_f32_16x16x4_f32(false, a, false, b1f, (short)0, acc1v, false, false);
    acc2v = __builtin_amdgcn_wmma_f32_16x16x4_f32(false, a, false, b2f, (short)0, acc2v, false, false);
  }

#pragma unroll
  for (int r = 0; r < 8; ++r) {
    const int row = rowBase + 8 * half + r;
    out[row * NCLASS + mrow]      = acc0[r];
    out[row * NCLASS + 16 + mrow] = acc1v[r];
    if (c2ok) out[row * NCLASS + 32 + mrow] = acc2v[r];
  }
}

// ---------------------------------------------------------------------------
// Zero fill (float4), COO edge scatter via native global_atomic_add_f32.
// ---------------------------------------------------------------------------
__global__ void zero4(float4* __restrict__ p, int n4) {
  const int i = blockIdx.x * blockDim.x + threadIdx.x;
  if (i < n4) p[i] = make_float4(0.f, 0.f, 0.f, 0.f);
}

__global__ void scatter64(const int* __restrict__ src, const int* __restrict__ dst,
                          const float* __restrict__ w, const float* __restrict__ feat,
                          float* __restrict__ acc) {
  const int tid = blockIdx.x * blockDim.x + threadIdx.x;
  if (tid >= N_EDGES * 16) return;
  const int e = tid >> 4;
  const int g = (tid & 15) * 4;
  const float ww = w[e];
  const float4 v = *(const float4*)(feat + src[e] * NHID + g);
  float* a = acc + dst[e] * NHID + g;
  unsafeAtomicAdd(a + 0, ww * v.x);
  unsafeAtomicAdd(a + 1, ww * v.y);
  unsafeAtomicAdd(a + 2, ww * v.z);
  unsafeAtomicAdd(a + 3, ww * v.w);
}

__global__ void scatter40(const int* __restrict__ src, const int* __restrict__ dst,
                          const float* __restrict__ w, const float* __restrict__ feat,
                          float* __restrict__ acc) {
  const unsigned tid = blockIdx.x * blockDim.x + threadIdx.x;
  if (tid >= (unsigned)(N_EDGES * 10)) return;
  const unsigned e = tid / 10u;
  const int g = (int)(tid - e * 10u) * 4;
  const float ww = w[e];
  const float4 v = *(const float4*)(feat + src[e] * NCLASS + g);
  float* a = acc + dst[e] * NCLASS + g;
  unsafeAtomicAdd(a + 0, ww * v.x);
  unsafeAtomicAdd(a + 1, ww * v.y);
  unsafeAtomicAdd(a + 2, ww * v.z);
  unsafeAtomicAdd(a + 3, ww * v.w);
}

// ---------------------------------------------------------------------------
// Fused tail: h2 = relu(acc2 + b2); logits = h2 @ Wp + bp; log_softmax.
// Wp/bp/b2 staged in LDS; 40x40 GEMV fully unrolled so hv[]/o[] stay in VGPRs.
// ---------------------------------------------------------------------------
__global__ __launch_bounds__(256) void predict_fused(const float* __restrict__ acc2,
                                                     const float* __restrict__ b2,
                                                     const float* __restrict__ Wp,
                                                     const float* __restrict__ bp,
                                                     float* __restrict__ out) {
  __shared__ float wS[NCLASS * NCLASS];
  __shared__ float bS[NCLASS];
  __shared__ float b2S[NCLASS];
  for (int i = threadIdx.x; i < NCLASS * NCLASS; i += blockDim.x) wS[i] = Wp[i];
  if (threadIdx.x < NCLASS) {
    bS[threadIdx.x]  = bp[threadIdx.x];
    b2S[threadIdx.x] = b2[threadIdx.x];
  }
  __syncthreads();

  const int r = blockIdx.x * blockDim.x + threadIdx.x;
  if (r >= N_NODES) return;

  float hv[NCLASS];
#pragma unroll
  for (int k = 0; k < NCLASS; ++k)
    hv[k] = fmaxf(acc2[r * NCLASS + k] + b2S[k], 0.f);

  float o[NCLASS];
#pragma unroll
  for (int c = 0; c < NCLASS; ++c) {
    float s = bS[c];
#pragma unroll
    for (int k = 0; k < NCLASS; ++k) s += hv[k] * wS[k * NCLASS + c];
    o[c] = s;
  }

  float m = o[0];
#pragma unroll
  for (int i = 1; i < NCLASS; ++i) m = fmaxf(m, o[i]);
  float s = 0.f;
#pragma unroll
  for (int i = 0; i < NCLASS; ++i) s += expf(o[i] - m);
  const float l = logf(s);
#pragma unroll
  for (int i = 0; i < NCLASS; ++i) out[r * NCLASS + i] = o[i] - m - l;
}

// ---------------------------------------------------------------------------
extern "C" void kernel_launch(void* const* d_in, const int* in_sizes, int n_in,
                              void* d_out, int out_size, void* d_ws, size_t ws_size,
                              hipStream_t stream) {
  const float* x  = (const float*)d_in[0];
  const int*   es = (const int*)d_in[1];
  const int*   ed = (const int*)d_in[2];
  const float* ew = (const float*)d_in[3];
  const float* W1 = (const float*)d_in[4];
  const float* b1 = (const float*)d_in[5];
  const float* W2 = (const float*)d_in[6];
  const float* b2 = (const float*)d_in[7];
  const float* Wp = (const float*)d_in[8];
  const float* bp = (const float*)d_in[9];
  float* out = (float*)d_out;

  float* bufA = (float*)d_ws;                 // 50000*64 f32
  float* bufB = bufA + N_NODES * NHID;        // 50000*64 f32  (25.6 MB total)

  const int gblk = ((N_NODES / 16) + 7) / 8;  // 8 waves (256 thr) per block
  const int n64  = N_NODES * NHID;
  const int n40  = N_NODES * NCLASS;

  // 1) XW = x @ W1                                   -> bufA
  gemm1_wmma<<<gblk, 256, NFEAT * NHID * sizeof(float), stream>>>(x, W1, bufA);
  // 2) acc1 = 0                                      (bufB)
  zero4<<<(n64 / 4 + 255) / 256, 256, 0, stream>>>((float4*)bufB, n64 / 4);
  // 3) acc1[dst] += w * XW[src]
  scatter64<<<(N_EDGES * 16 + 255) / 256, 256, 0, stream>>>(es, ed, ew, bufA, bufB);
  // 4) HW = relu(acc1 + b1) @ W2                     -> bufA
  gemm2_wmma<<<gblk, 256, 0, stream>>>(bufB, b1, W2, bufA);
  // 5) acc2 = 0                                      (bufB)
  zero4<<<(n40 / 4 + 255) / 256, 256, 0, stream>>>((float4*)bufB, n40 / 4);
  // 6) acc2[dst] += w * HW[src]
  scatter40<<<(N_EDGES * 10 + 255) / 256, 256, 0, stream>>>(es, ed, ew, bufA, bufB);
  // 7) out = log_softmax(relu(acc2 + b2) @ Wp + bp)  -> d_out
  predict_fused<<<(N_NODES + 255) / 256, 256, 0, stream>>>(bufB, b2, Wp, bp, out);
}